// MidMTG_63711544869372
// MI455X (gfx1250) — compile-verified
//
#include <hip/hip_runtime.h>

typedef __attribute__((ext_vector_type(2))) float v2f;
typedef __attribute__((ext_vector_type(8))) float v8f;

#define B_ROWS   4096
#define NC       1024
#define NL       512
#define GAMMA_C  0.95f
#define KAPPA_C  0.1f
#define T_STEPS  50
#define CONF_FLOOR 1e-6f
#define LDSTRIDE 516   // 512 + 4: breaks 2-way bank conflict between half-waves

__global__ __launch_bounds__(512)
void MidMTG_lemma_kernel(const float* __restrict__ c_lex,
                         const float* __restrict__ W,
                         const float* __restrict__ a0,
                         float* __restrict__ out) {
    __shared__ float ldsD[16 * LDSTRIDE];

    const int tid  = threadIdx.x;
    const int wave = tid >> 5;        // 0..15
    const int lane = tid & 31;
    const int r    = lane & 15;       // M index (A) / N index (B,C,D) within tile
    const int kh   = lane >> 4;       // K-half selector for A/B fragments
    const int row0 = blockIdx.x * 16; // this block's 16 rows of B

    // ---------------- Phase 1: D = c_lex @ W^T via f32 WMMA ----------------
    // Each wave computes TWO adjacent 16x16 N-tiles in one K-loop so the A
    // fragment (c_lex) is loaded once and fed to both WMMAs: 3 loads / 2 wmma.
    // A 16x4 f32 fragment: lane holds A[M=r][K=2*kh .. 2*kh+1]   (contiguous)
    // B 4x16 f32 fragment: lane holds B[K=2*kh .. 2*kh+1][N=r],
    //                      B[k][n] = W[n][k] -> contiguous along W's row
    // D/C 16x16 f32: VGPR v of lane = D[M = v + 8*kh][N = r]
    {
        const int col0 = wave * 32;                 // first of the two N-tiles
        const float* pa  = c_lex + (size_t)(row0 + r) * NC + 2 * kh;
        const float* pb0 = W     + (size_t)(col0 + r)      * NC + 2 * kh;
        const float* pb1 = pb0   + (size_t)16 * NC;          // second N-tile
        v8f acc0 = {};
        v8f acc1 = {};
        #pragma unroll 4
        for (int k = 0; k < NC; k += 4) {
            v2f av = *(const v2f*)(pa  + k);
            v2f b0 = *(const v2f*)(pb0 + k);
            v2f b1 = *(const v2f*)(pb1 + k);
            acc0 = __builtin_amdgcn_wmma_f32_16x16x4_f32(
                false, av, false, b0, (short)0, acc0, false, false);
            acc1 = __builtin_amdgcn_wmma_f32_16x16x4_f32(
                false, av, false, b1, (short)0, acc1, false, false);
        }
        #pragma unroll
        for (int v = 0; v < 8; ++v) {
            ldsD[(v + 8 * kh) * LDSTRIDE + col0 + r]      = acc0[v];
            ldsD[(v + 8 * kh) * LDSTRIDE + col0 + 16 + r] = acc1[v];
        }
    }
    __syncthreads();

    // ---------------- Phase 2: 50-cycle recurrence, wave w == row w --------
    const int row = row0 + wave;
    float d[16], a[16];
    #pragma unroll
    for (int j = 0; j < 16; ++j) {
        const int col = lane + 32 * j;
        d[j] = ldsD[wave * LDSTRIDE + col];
        a[j] = a0[(size_t)row * NL + col];
    }

    for (int s = 0; s < T_STEPS; ++s) {
        float part = 0.f;
        #pragma unroll
        for (int j = 0; j < 16; ++j) {
            a[j] = fmaf(GAMMA_C, a[j], d[j]);   // decay + drive
            part += a[j];
        }
        // rowsum across the full 512 lemmas: 16 local + wave32 tree reduction
        #pragma unroll
        for (int off = 16; off >= 1; off >>= 1)
            part += __shfl_xor(part, off, 32);
        #pragma unroll
        for (int j = 0; j < 16; ++j)
            a[j] = fmaxf(fmaf(-KAPPA_C, part, (1.0f + KAPPA_C) * a[j]), 0.0f);
    }

    // ---------------- Phase 3: outputs (a, selected, confidence) -----------
    float* out_a    = out;
    float* out_sel  = out + (size_t)B_ROWS * NL;
    float* out_conf = out_sel + B_ROWS;

    float sum = 0.f, bv = -1.0f;
    int   bi  = 0;
    #pragma unroll
    for (int j = 0; j < 16; ++j) {
        const int col = lane + 32 * j;
        out_a[(size_t)row * NL + col] = a[j];
        sum += a[j];
        if (a[j] > bv) { bv = a[j]; bi = col; }   // first-max within lane (cols ascend)
    }
    #pragma unroll
    for (int off = 16; off >= 1; off >>= 1) {
        sum += __shfl_xor(sum, off, 32);
        float ov = __shfl_xor(bv, off, 32);
        int   oi = __shfl_xor(bi, off, 32);
        if (ov > bv || (ov == bv && oi < bi)) { bv = ov; bi = oi; }  // JAX first-max tie-break
    }
    if (lane == 0) {
        const float mean = sum * (1.0f / (float)NL);
        out_sel[row]  = (float)bi;
        out_conf[row] = bv / fmaxf(mean, CONF_FLOOR);
    }
}

extern "C" void kernel_launch(void* const* d_in, const int* in_sizes, int n_in,
                              void* d_out, int out_size, void* d_ws, size_t ws_size,
                              hipStream_t stream) {
    (void)in_sizes; (void)n_in; (void)out_size; (void)d_ws; (void)ws_size;
    const float* c_lex = (const float*)d_in[0];
    const float* W     = (const float*)d_in[1];
    const float* a0    = (const float*)d_in[2];
    float* out         = (float*)d_out;

    dim3 grid(B_ROWS / 16);   // 256 blocks
    dim3 block(512);          // 16 waves (wave32)
    MidMTG_lemma_kernel<<<grid, block, 0, stream>>>(c_lex, W, a0, out);
}